// GCN_13030930776648
// MI455X (gfx1250) — compile-verified
//
#include <hip/hip_runtime.h>

typedef float v2f __attribute__((ext_vector_type(2)));
typedef float v8f __attribute__((ext_vector_type(8)));

// ---------------- zero scratch (sums + counts) ----------------
__global__ void rgcn_zero_u32(unsigned* __restrict__ p, int n) {
  int i = blockIdx.x * blockDim.x + threadIdx.x;
  int st = gridDim.x * blockDim.x;
  for (; i < n; i += st) p[i] = 0u;
}

// ---------------- root transform via V_WMMA_F32_16X16X4_F32 ----------------
// tmp[n][c] = sum_k in[n][k]*root[k][c] + bias[c] ; 16 nodes per wave.
// A = root^T (16x4 MxK, rows m=0..1 live): lanes 0-15 hold K=0,1 ; 16-31 K=2,3
// B = x^T tile (4x16 KxN, N=node):        lanes 0-15 hold K=0,1 ; 16-31 K=2,3
// D (16x16): reg r, lanes 0-15 -> M=r, N=lane  => chan0 in reg0, chan1 in reg1,
// so lanes 0-15 emit one coalesced float2 store per node.
template <int CIN>
__global__ void rgcn_root_wmma(const float* __restrict__ in,
                               const float* __restrict__ root,   // [CIN][2]
                               const float* __restrict__ bias,   // [2]
                               float* __restrict__ tmp, int N, int nTiles) {
  int gid  = blockIdx.x * blockDim.x + threadIdx.x;
  int wave = gid >> 5;
  int lane = threadIdx.x & 31;
  if (wave >= nTiles) return;          // wave-uniform exit: EXEC all-1s at WMMA
  int base = wave * 16;
  int col  = lane & 15;

  v2f a; a.x = 0.f; a.y = 0.f;         // A = root^T
  if (col < 2) {
    if (lane < 16) {
      a.x = root[0 * 2 + col];                       // K=0
      if (CIN > 1) a.y = root[1 * 2 + col];          // K=1
    } else {
      if (CIN > 2) a.x = root[2 * 2 + col];          // K=2
      if (CIN > 3) a.y = root[3 * 2 + col];          // K=3
    }
  }

  v2f b; b.x = 0.f; b.y = 0.f;         // B = x^T tile, N = node
  int node = base + col;
  bool ok = node < N;
  if (lane < 16) {
    if (ok) {
      if (CIN == 2) {
        float2 v = ((const float2*)in)[node];        // b64 gather
        b.x = v.x; b.y = v.y;
      } else {
        b.x = in[node * CIN + 0];
        if (CIN > 1) b.y = in[node * CIN + 1];
      }
    }
  } else {
    if (ok && CIN > 2) {
      b.x = in[node * CIN + 2];
      if (CIN > 3) b.y = in[node * CIN + 3];
    }
  }

  v8f c = {};
  c = __builtin_amdgcn_wmma_f32_16x16x4_f32(false, a, false, b,
                                            (short)0, c, false, false);

  if (lane < 16 && node < N) {         // coalesced float2 store, 16 lanes
    float2 o;
    o.x = c[0] + bias[0];
    o.y = c[1] + bias[1];
    ((float2*)tmp)[node] = o;
  }
}

// ---------------- per-edge work, layer 1 ----------------
__device__ __forceinline__ unsigned long long
rgcn_edge1_one(int s, int d, int r, const float* __restrict__ x,
               const float* __restrict__ w,          // 18 regs after inlining
               float* __restrict__ sums, int* __restrict__ cnt) {
  float x0 = x[3 * s], x1 = x[3 * s + 1], x2 = x[3 * s + 2];
  float w00 = (r == 0) ? w[0] : (r == 1) ? w[6]  : w[12];
  float w01 = (r == 0) ? w[1] : (r == 1) ? w[7]  : w[13];
  float w10 = (r == 0) ? w[2] : (r == 1) ? w[8]  : w[14];
  float w11 = (r == 0) ? w[3] : (r == 1) ? w[9]  : w[15];
  float w20 = (r == 0) ? w[4] : (r == 1) ? w[10] : w[16];
  float w21 = (r == 0) ? w[5] : (r == 1) ? w[11] : w[17];
  float m0 = x0 * w00 + x1 * w10 + x2 * w20;
  float m1 = x0 * w01 + x1 * w11 + x2 * w21;
  float* sp = sums + (size_t)d * 6 + (size_t)r * 2;
  atomicAdd(sp, m0);
  atomicAdd(sp + 1, m1);
  atomicAdd(cnt + d * 3 + r, 1);
  return (unsigned long long)(unsigned)s |
         ((unsigned long long)(unsigned)d << 20) |
         ((unsigned long long)(unsigned)r << 40);
}

// layer-1, 4 edges/iteration via b128 loads (requires E % 4 == 0)
__global__ void rgcn_edge1_vec4(const int4* __restrict__ src4,
                                const int4* __restrict__ dst4,
                                const int4* __restrict__ rel4,
                                const float* __restrict__ x,
                                const float* __restrict__ W,      // [3][3][2]
                                float* __restrict__ sums, int* __restrict__ cnt,
                                ulonglong2* __restrict__ packed2,
                                int usePacked, int Q) {
  float w[18];
#pragma unroll
  for (int i = 0; i < 18; ++i) w[i] = W[i];
  int i  = blockIdx.x * blockDim.x + threadIdx.x;
  int st = gridDim.x * blockDim.x;
  for (; i < Q; i += st) {
    int4 s = src4[i], d = dst4[i], r = rel4[i];
    unsigned long long p0 = rgcn_edge1_one(s.x, d.x, r.x, x, w, sums, cnt);
    unsigned long long p1 = rgcn_edge1_one(s.y, d.y, r.y, x, w, sums, cnt);
    unsigned long long p2 = rgcn_edge1_one(s.z, d.z, r.z, x, w, sums, cnt);
    unsigned long long p3 = rgcn_edge1_one(s.w, d.w, r.w, x, w, sums, cnt);
    if (usePacked) {
      ulonglong2 a; a.x = p0; a.y = p1;
      ulonglong2 b; b.x = p2; b.y = p3;
      packed2[2 * i]     = a;                        // b128 stores
      packed2[2 * i + 1] = b;
    }
  }
}

// layer-1 scalar fallback
__global__ void rgcn_edge1(const int* __restrict__ src, const int* __restrict__ dst,
                           const int* __restrict__ rel, const float* __restrict__ x,
                           const float* __restrict__ W,
                           float* __restrict__ sums, int* __restrict__ cnt,
                           unsigned long long* __restrict__ packed, int usePacked,
                           int E) {
  float w[18];
#pragma unroll
  for (int i = 0; i < 18; ++i) w[i] = W[i];
  int i  = blockIdx.x * blockDim.x + threadIdx.x;
  int st = gridDim.x * blockDim.x;
  for (; i < E; i += st) {
    unsigned long long p = rgcn_edge1_one(src[i], dst[i], rel[i], x, w, sums, cnt);
    if (usePacked) packed[i] = p;
  }
}

// ---------------- per-edge work, layer 2 ----------------
__device__ __forceinline__ void
rgcn_edge2_one(int s, int d, int r, const float* __restrict__ h,
               const float* __restrict__ w, float* __restrict__ sums) {
  float2 hv = ((const float2*)h)[s];                 // b64 gather
  float w00 = (r == 0) ? w[0] : (r == 1) ? w[4] : w[8];
  float w01 = (r == 0) ? w[1] : (r == 1) ? w[5] : w[9];
  float w10 = (r == 0) ? w[2] : (r == 1) ? w[6] : w[10];
  float w11 = (r == 0) ? w[3] : (r == 1) ? w[7] : w[11];
  float m0 = hv.x * w00 + hv.y * w10;
  float m1 = hv.x * w01 + hv.y * w11;
  float* sp = sums + (size_t)d * 6 + (size_t)r * 2;
  atomicAdd(sp, m0);
  atomicAdd(sp + 1, m1);
}

__device__ __forceinline__ void
rgcn_edge2_unpack(unsigned long long p, const float* __restrict__ h,
                  const float* __restrict__ w, float* __restrict__ sums) {
  int s = (int)(p & 0xFFFFFull);
  int d = (int)((p >> 20) & 0xFFFFFull);
  int r = (int)(p >> 40);
  rgcn_edge2_one(s, d, r, h, w, sums);
}

// layer-2, 2 packed edges/iteration via b128 loads (requires E % 2 == 0)
__global__ void rgcn_edge2_vec2(const ulonglong2* __restrict__ packed2,
                                const float* __restrict__ h,
                                const float* __restrict__ W,      // [3][2][2]
                                float* __restrict__ sums, int H) {
  float w[12];
#pragma unroll
  for (int i = 0; i < 12; ++i) w[i] = W[i];
  int i  = blockIdx.x * blockDim.x + threadIdx.x;
  int st = gridDim.x * blockDim.x;
  for (; i < H; i += st) {
    ulonglong2 p = packed2[i];
    rgcn_edge2_unpack(p.x, h, w, sums);
    rgcn_edge2_unpack(p.y, h, w, sums);
  }
}

// layer-2 scalar packed fallback
__global__ void rgcn_edge2_packed(const unsigned long long* __restrict__ packed,
                                  const float* __restrict__ h,
                                  const float* __restrict__ W,
                                  float* __restrict__ sums, int E) {
  float w[12];
#pragma unroll
  for (int i = 0; i < 12; ++i) w[i] = W[i];
  int i  = blockIdx.x * blockDim.x + threadIdx.x;
  int st = gridDim.x * blockDim.x;
  for (; i < E; i += st) rgcn_edge2_unpack(packed[i], h, w, sums);
}

// layer-2 raw fallback (no packed workspace)
__global__ void rgcn_edge2_raw(const int* __restrict__ src, const int* __restrict__ dst,
                               const int* __restrict__ rel,
                               const float* __restrict__ h,
                               const float* __restrict__ W,
                               float* __restrict__ sums, int E) {
  float w[12];
#pragma unroll
  for (int i = 0; i < 12; ++i) w[i] = W[i];
  int i  = blockIdx.x * blockDim.x + threadIdx.x;
  int st = gridDim.x * blockDim.x;
  for (; i < E; i += st) rgcn_edge2_one(src[i], dst[i], rel[i], h, w, sums);
}

// ---------------- finalize: mean over relations + root term (+ReLU) --------
__global__ void rgcn_finalize(const float* __restrict__ tmp,
                              const float* __restrict__ sums,
                              const int* __restrict__ cnt,
                              float* __restrict__ out, int N, int doRelu) {
  int n  = blockIdx.x * blockDim.x + threadIdx.x;
  int st = gridDim.x * blockDim.x;
  for (; n < N; n += st) {
    float2 t = ((const float2*)tmp)[n];
    float a0 = t.x, a1 = t.y;
    const float* sp = sums + (size_t)n * 6;
#pragma unroll
    for (int r = 0; r < 3; ++r) {
      float k = fmaxf((float)cnt[n * 3 + r], 1.0f);
      float2 sv = ((const float2*)sp)[r];
      a0 += sv.x / k;
      a1 += sv.y / k;
    }
    if (doRelu) { a0 = fmaxf(a0, 0.f); a1 = fmaxf(a1, 0.f); }
    float2 o; o.x = a0; o.y = a1;
    ((float2*)out)[n] = o;
  }
}

extern "C" void kernel_launch(void* const* d_in, const int* in_sizes, int n_in,
                              void* d_out, int out_size, void* d_ws, size_t ws_size,
                              hipStream_t stream) {
  const float* x     = (const float*)d_in[0];
  const int*   eidx  = (const int*)d_in[1];
  const int*   eattr = (const int*)d_in[2];
  const float* W1    = (const float*)d_in[3];
  const float* root1 = (const float*)d_in[4];
  const float* b1    = (const float*)d_in[5];
  const float* W2    = (const float*)d_in[6];
  const float* root2 = (const float*)d_in[7];
  const float* b2    = (const float*)d_in[8];

  const int E = in_sizes[2];          // 16M edges
  const int N = in_sizes[0] / 3;      // 1M nodes
  const int* src = eidx;
  const int* dst = eidx + E;

  // workspace layout (sums and cnt contiguous: one zero pass covers both)
  char* ws = (char*)d_ws;
  size_t offTmp    = 0;
  size_t offH      = offTmp  + (size_t)N * 2 * sizeof(float);
  size_t offSums   = offH    + (size_t)N * 2 * sizeof(float);
  size_t offCnt    = offSums + (size_t)N * 6 * sizeof(float);
  size_t offPacked = offCnt  + (size_t)N * 3 * sizeof(int);
  offPacked = (offPacked + 15) & ~(size_t)15;

  float* tmp  = (float*)(ws + offTmp);
  float* h    = (float*)(ws + offH);
  float* sums = (float*)(ws + offSums);
  int*   cnt  = (int*)  (ws + offCnt);
  unsigned long long* packed = (unsigned long long*)(ws + offPacked);
  const int usePacked = (N <= (1 << 20) &&
      ws_size >= offPacked + (size_t)E * sizeof(unsigned long long)) ? 1 : 0;
  const int vec4 = ((E & 3) == 0) ? 1 : 0;   // b128 path needs E % 4 == 0

  const int tiles      = (N + 15) / 16;
  const int rootBlocks = (tiles + 7) / 8;    // 8 waves per 256-thread block
  const int edgeBlocks = 4096;
  const int nodeBlocks = (N + 255) / 256;

  // ---- layer 1 ----
  rgcn_zero_u32<<<1024, 256, 0, stream>>>((unsigned*)sums, N * 9);  // sums+cnt
  rgcn_root_wmma<3><<<rootBlocks, 256, 0, stream>>>(x, root1, b1, tmp, N, tiles);
  if (vec4)
    rgcn_edge1_vec4<<<edgeBlocks, 256, 0, stream>>>(
        (const int4*)src, (const int4*)dst, (const int4*)eattr, x, W1,
        sums, cnt, (ulonglong2*)packed, usePacked, E / 4);
  else
    rgcn_edge1<<<edgeBlocks, 256, 0, stream>>>(src, dst, eattr, x, W1,
                                               sums, cnt, packed, usePacked, E);
  rgcn_finalize<<<nodeBlocks, 256, 0, stream>>>(tmp, sums, cnt, h, N, 1);

  // ---- layer 2 (counts reused; only sums re-zeroed) ----
  rgcn_zero_u32<<<1024, 256, 0, stream>>>((unsigned*)sums, N * 6);
  rgcn_root_wmma<2><<<rootBlocks, 256, 0, stream>>>(h, root2, b2, tmp, N, tiles);
  if (usePacked) {
    if ((E & 1) == 0)
      rgcn_edge2_vec2<<<edgeBlocks, 256, 0, stream>>>(
          (const ulonglong2*)packed, h, W2, sums, E / 2);
    else
      rgcn_edge2_packed<<<edgeBlocks, 256, 0, stream>>>(packed, h, W2, sums, E);
  } else {
    rgcn_edge2_raw<<<edgeBlocks, 256, 0, stream>>>(src, dst, eattr, h, W2, sums, E);
  }
  rgcn_finalize<<<nodeBlocks, 256, 0, stream>>>(tmp, sums, cnt, (float*)d_out, N, 0);
}